// GAT_49168785605117
// MI455X (gfx1250) — compile-verified
//
#include <hip/hip_runtime.h>
#include <hip/hip_bf16.h>
#include <math.h>

typedef __attribute__((ext_vector_type(2))) float v2f;
typedef __attribute__((ext_vector_type(8))) float v8f;

#define N_NODES 50000
#define NEG_SLOPE 0.2f

// ---------------------------------------------------------------------------
// Monotonic float <-> unsigned key (order-preserving), so segment-max can be a
// single u32 atomicMax pass. Key 0 == "smaller than every real float".
// ---------------------------------------------------------------------------
__device__ __forceinline__ unsigned fkey(float x) {
  unsigned b = __float_as_uint(x);
  return (b & 0x80000000u) ? ~b : (b | 0x80000000u);
}
__device__ __forceinline__ float funkey(unsigned k) {
  unsigned b = (k & 0x80000000u) ? (k & 0x7FFFFFFFu) : ~k;
  return __uint_as_float(b);
}

// ---------------------------------------------------------------------------
// Zero-fill scratch (graph-capture-safe replacement for hipMemsetAsync)
// ---------------------------------------------------------------------------
__global__ void gat_zero_f32(float* __restrict__ p, int n) {
  int i = blockIdx.x * blockDim.x + threadIdx.x;
  if (i < n) p[i] = 0.0f;
}

// ---------------------------------------------------------------------------
// H = X @ W via V_WMMA_F32_16X16X4_F32, compile-time (Fin, Fout).
// One wave owns a 16-row strip and ALL NT = ceil(Fout/16) column tiles:
// A fragment loaded once per k-step, reused across NT WMMAs (NT accumulators).
//   A (16x4 f32): lane<16 -> row M=lane, v0/v1 = K0,K1 ; lane>=16 -> K2,K3
//   B (4x16 f32): lane<16 -> col N=lane, v0/v1 = K0,K1 ; lane>=16 -> K2,K3
//   C/D: VGPR r holds M = r + 8*(lane>=16), N = lane&15
// Rows guarded per-wave (uniform exit, EXEC stays all-ones at every WMMA).
// Tail columns (Fout=40) handled by clamped-index + mask multiply (v_cndmask,
// no EXEC manipulation); full tiles fold the mask away at compile time.
// ---------------------------------------------------------------------------
template <int Fin, int Fout>
__global__ __launch_bounds__(128) void gat_gemm_wmma(
    const float* __restrict__ X, const float* __restrict__ W,
    float* __restrict__ H, int nStrips) {
  constexpr int NT = (Fout + 15) / 16;
  const int strip = blockIdx.x * blockDim.y + threadIdx.y;  // wave-uniform
  if (strip >= nStrips) return;                              // whole-wave exit
  const int lane = threadIdx.x;  // 0..31
  const int half = lane >> 4;
  const int l16  = lane & 15;

  const float* xrow = X + (size_t)(strip * 16 + l16) * Fin;

  v8f acc[NT];
#pragma unroll
  for (int t = 0; t < NT; ++t) acc[t] = (v8f){};

  for (int k = 0; k < Fin; k += 4) {
    const int ka = k + 2 * half;
    v2f a;
    a.x = xrow[ka];
    a.y = xrow[ka + 1];
#pragma unroll
    for (int t = 0; t < NT; ++t) {
      const int col = t * 16 + l16;
      v2f b;
      if ((t + 1) * 16 <= Fout) {          // full tile: unconditional loads
        b.x = W[ka * Fout + col];
        b.y = W[(ka + 1) * Fout + col];
      } else {                              // tail tile: clamp + mask
        const int  cc  = (col < Fout) ? col : (Fout - 1);
        const float mk = (col < Fout) ? 1.0f : 0.0f;
        b.x = W[ka * Fout + cc] * mk;
        b.y = W[(ka + 1) * Fout + cc] * mk;
      }
      acc[t] = __builtin_amdgcn_wmma_f32_16x16x4_f32(
          /*neg_a=*/false, a, /*neg_b=*/false, b,
          /*c_mod=*/(short)0, acc[t], /*reuse_a=*/false, /*reuse_b=*/false);
    }
  }

  const int rbase = strip * 16 + 8 * half;
#pragma unroll
  for (int t = 0; t < NT; ++t) {
    const int col = t * 16 + l16;
    if ((t + 1) * 16 <= Fout) {
#pragma unroll
      for (int r = 0; r < 8; ++r)
        H[(size_t)(rbase + r) * Fout + col] = acc[t][r];
    } else if (col < Fout) {
#pragma unroll
      for (int r = 0; r < 8; ++r)
        H[(size_t)(rbase + r) * Fout + col] = acc[t][r];
    }
  }
}

// ---------------------------------------------------------------------------
// s[n] = h[n,:].a_src ; d[n] = h[n,:].a_dst  (wave-per-node, coalesced)
// ---------------------------------------------------------------------------
__global__ void gat_attn_terms(const float* __restrict__ H,
                               const float* __restrict__ as,
                               const float* __restrict__ ad,
                               float* __restrict__ s, float* __restrict__ d,
                               int N, int F) {
  const int w = (blockIdx.x * blockDim.x + threadIdx.x) >> 5;
  const int lane = threadIdx.x & 31;
  if (w >= N) return;
  const float* row = H + (size_t)w * F;
  float ss = 0.0f, dd = 0.0f;
  for (int f = lane; f < F; f += 32) {
    const float h = row[f];
    ss += h * as[f];
    dd += h * ad[f];
  }
  for (int o = 16; o > 0; o >>= 1) {
    ss += __shfl_xor(ss, o, 32);
    dd += __shfl_xor(dd, o, 32);
  }
  if (lane == 0) { s[w] = ss; d[w] = dd; }
}

// Edge j in [0, E+N): j<E from edge_index, else self-loop j-E.
__device__ __forceinline__ void edge_sd(const long long* __restrict__ ei,
                                        int j, int E, int& sj, int& dj) {
  if (j < E) { sj = (int)ei[j]; dj = (int)ei[(size_t)E + j]; }
  else       { sj = dj = j - E; }
}

// e[j] = leaky_relu(s[src]+d[dst]); segment-max into mkey[dst]
__global__ void gat_edge_logits(const long long* __restrict__ ei,
                                const float* __restrict__ s,
                                const float* __restrict__ d,
                                float* __restrict__ e,
                                unsigned* __restrict__ mkey,
                                int E, int N) {
  const int j = blockIdx.x * blockDim.x + threadIdx.x;
  if (j >= E + N) return;
  int sj, dj; edge_sd(ei, j, E, sj, dj);
  float v = s[sj] + d[dj];
  v = (v > 0.0f) ? v : NEG_SLOPE * v;
  e[j] = v;
  atomicMax(&mkey[dj], fkey(v));
}

// p[j] = exp(e[j]-m[dst]) (in place); denom[dst] += p
__global__ void gat_edge_exp(const long long* __restrict__ ei,
                             float* __restrict__ e,
                             const unsigned* __restrict__ mkey,
                             float* __restrict__ denom,
                             int E, int N) {
  const int j = blockIdx.x * blockDim.x + threadIdx.x;
  if (j >= E + N) return;
  int sj, dj; edge_sd(ei, j, E, sj, dj);
  const float p = __expf(e[j] - funkey(mkey[dj]));
  e[j] = p;
  atomicAdd(&denom[dj], p);
}

// agg[dst,:] += (p[j]/denom[dst]) * h[src,:]  — one wave per edge,
// lanes stride features (coalesced 128B bursts, hits L2).
__global__ void gat_scatter(const long long* __restrict__ ei,
                            const float* __restrict__ p,
                            const float* __restrict__ denom,
                            const float* __restrict__ H,
                            float* __restrict__ agg,
                            int E, int N, int F) {
  const int j = blockIdx.x * blockDim.y + threadIdx.y;  // edge id (wave-uniform)
  const int lane = threadIdx.x;                          // 0..31
  if (j >= E + N) return;
  int sj, dj; edge_sd(ei, j, E, sj, dj);
  const float alpha = p[j] / denom[dj];
  const float* hrow = H + (size_t)sj * F;
  float* orow = agg + (size_t)dj * F;
  for (int f = lane; f < F; f += 32)
    atomicAdd(&orow[f], alpha * hrow[f]);
}

// y = agg + b ; out = y*sigmoid(y)  (SiLU)
__global__ void gat_bias_silu(const float* __restrict__ agg,
                              const float* __restrict__ b,
                              float* __restrict__ out, int total, int F) {
  const int i = blockIdx.x * blockDim.x + threadIdx.x;
  if (i >= total) return;
  const float y = agg[i] + b[i % F];
  out[i] = y / (1.0f + __expf(-y));
}

// final layer: y = agg + b ; out = log_softmax(y) along F (wave-per-node)
__global__ void gat_bias_logsoftmax(const float* __restrict__ agg,
                                    const float* __restrict__ b,
                                    float* __restrict__ out, int N, int F) {
  const int w = (blockIdx.x * blockDim.x + threadIdx.x) >> 5;
  const int lane = threadIdx.x & 31;
  if (w >= N) return;
  const float* row = agg + (size_t)w * F;
  const int f0 = lane, f1 = lane + 32;
  float y0 = (f0 < F) ? row[f0] + b[f0] : -INFINITY;
  float y1 = (f1 < F) ? row[f1] + b[f1] : -INFINITY;
  float mx = fmaxf(y0, y1);
  for (int o = 16; o > 0; o >>= 1) mx = fmaxf(mx, __shfl_xor(mx, o, 32));
  float sm = ((f0 < F) ? __expf(y0 - mx) : 0.0f) +
             ((f1 < F) ? __expf(y1 - mx) : 0.0f);
  for (int o = 16; o > 0; o >>= 1) sm += __shfl_xor(sm, o, 32);
  const float ls = __logf(sm);
  if (f0 < F) out[(size_t)w * F + f0] = y0 - mx - ls;
  if (f1 < F) out[(size_t)w * F + f1] = y1 - mx - ls;
}

// ---------------------------------------------------------------------------
extern "C" void kernel_launch(void* const* d_in, const int* in_sizes, int n_in,
                              void* d_out, int out_size, void* d_ws, size_t ws_size,
                              hipStream_t stream) {
  const float*     x  = (const float*)d_in[0];
  const long long* ei = (const long long*)d_in[1];   // int64 [2,E]
  const float* W[3]  = {(const float*)d_in[2], (const float*)d_in[6], (const float*)d_in[10]};
  const float* as[3] = {(const float*)d_in[3], (const float*)d_in[7], (const float*)d_in[11]};
  const float* ad[3] = {(const float*)d_in[4], (const float*)d_in[8], (const float*)d_in[12]};
  const float* bb[3] = {(const float*)d_in[5], (const float*)d_in[9], (const float*)d_in[13]};

  const int N = N_NODES;
  const int E = in_sizes[1] / 2;
  const int EA = E + N;                     // edges incl. self-loops
  const int dims[4] = {128, 128, 64, 40};

  // workspace layout (floats): hbuf | xbuf | agg | mkey | denom | svec | dvec | evec
  float* ws     = (float*)d_ws;
  float* hbuf   = ws;                              // N*128
  float* xbuf   = hbuf + (size_t)N * 128;          // N*128
  float* agg    = xbuf + (size_t)N * 128;          // N*128
  unsigned* mkey= (unsigned*)(agg + (size_t)N * 128); // N  (contiguous after agg)
  float* denom  = (float*)mkey + N;                // N
  float* svec   = denom + N;                       // N
  float* dvec   = svec + N;                        // N
  float* evec   = dvec + N;                        // E+N
  (void)ws_size; (void)n_in; (void)out_size;

  const int nStrips = N / 16;        // 3125, exact
  const dim3 gblk(32, 4);            // 4 waves per block, one strip each
  const int gblocks = (nStrips + 3) / 4;

  const float* cur = x;
  for (int l = 0; l < 3; ++l) {
    const int Fin = dims[l], Fout = dims[l + 1];

    // zero agg (full 128-wide) + mkey + denom in one pass (contiguous)
    const int zn = N * 128 + 2 * N;
    gat_zero_f32<<<(zn + 255) / 256, 256, 0, stream>>>(agg, zn);

    // dense GEMM on the WMMA path (fp32, full precision)
    if (l == 0)
      gat_gemm_wmma<128, 128><<<gblocks, gblk, 0, stream>>>(cur, W[l], hbuf, nStrips);
    else if (l == 1)
      gat_gemm_wmma<128, 64><<<gblocks, gblk, 0, stream>>>(cur, W[l], hbuf, nStrips);
    else
      gat_gemm_wmma<64, 40><<<gblocks, gblk, 0, stream>>>(cur, W[l], hbuf, nStrips);

    // per-node attention terms
    gat_attn_terms<<<(N * 32 + 255) / 256, 256, 0, stream>>>(
        hbuf, as[l], ad[l], svec, dvec, N, Fout);

    // edge logits + segment max
    gat_edge_logits<<<(EA + 255) / 256, 256, 0, stream>>>(
        ei, svec, dvec, evec, mkey, E, N);

    // exp + segment sum
    gat_edge_exp<<<(EA + 255) / 256, 256, 0, stream>>>(
        ei, evec, mkey, denom, E, N);

    // weighted scatter: wave per edge, lanes over features
    dim3 sblk(32, 8);
    gat_scatter<<<(EA + 7) / 8, sblk, 0, stream>>>(
        ei, evec, denom, hbuf, agg, E, N, Fout);

    if (l < 2) {
      const int total = N * Fout;
      gat_bias_silu<<<(total + 255) / 256, 256, 0, stream>>>(
          agg, bb[l], xbuf, total, Fout);
      cur = xbuf;
    } else {
      gat_bias_logsoftmax<<<(N * 32 + 255) / 256, 256, 0, stream>>>(
          agg, bb[l], (float*)d_out, N, Fout);
    }
  }
}